// Memory_6468220747783
// MI455X (gfx1250) — compile-verified
//
#include <hip/hip_runtime.h>
#include <hip/hip_bf16.h>

typedef __attribute__((ext_vector_type(2))) float v2f;
typedef __attribute__((ext_vector_type(8))) float v8f;

#define BATCH  64
#define DIM    256
#define NMEM   100000
#define KTOP   128
#define VSZ    256
#define KEYSZ  32
#define QKVSZ  320
#define LN_EPS 1e-5f

static __device__ __forceinline__ unsigned f2key(float f) {
  unsigned u = __float_as_uint(f);
  return (u & 0x80000000u) ? ~u : (u | 0x80000000u);
}

// ---------------------------------------------------------------------------
// f32 WMMA GEMM: C[M,N] = act(A[M,K] @ W + bias)
// W element (k,n) at W[k*wsK + n*wsN]  (row-major: wsK=N,wsN=1; transposed
// row-major source: wsK=1, wsN=rowlen).
// Block = 256 threads = 8 waves; block tile 64(M) x 128(N). Each wave owns a
// 64x16 strip: one B fragment feeds FOUR v_wmma_f32_16x16x4_f32 ops, so the
// streaming W operand (mem_c for the 102MB similarity GEMM) is loaded exactly
// once per block -> minimal HBM traffic on the bandwidth-bound stage.
// Requires M % 64 == 0 (true for all call sites: 64, 8192).
// ---------------------------------------------------------------------------
__global__ void gemm_wmma_f32(const float* __restrict__ A, long lda,
                              const float* __restrict__ W, long wsK, long wsN,
                              const float* __restrict__ bias,
                              float* __restrict__ C, long ldc,
                              int M, int N, int K, int relu)
{
  const int wave = threadIdx.x >> 5;
  const int lane = threadIdx.x & 31;
  const int m0 = blockIdx.y * 64;
  const int n0 = blockIdx.x * 128 + wave * 16;
  if (n0 >= N) return;                             // wave-uniform
  const int half = lane >> 4;                      // 0 or 1
  const int lm   = lane & 15;
  const int kb   = half * 2;                       // K offset within 4-step
  const long bn  = n0 + lm;                        // W col this lane feeds
  const float* Abase = A + (long)(m0 + lm) * lda + kb;
  const long aRowStep = 16 * lda;

  v8f acc0 = {}, acc1 = {}, acc2 = {}, acc3 = {};
  for (int kk = 0; kk < K; kk += 4) {
    const float* wp = W + (long)(kk + kb) * wsK + bn * wsN;
    v2f b;
    b.x = wp[0]; b.y = wp[wsK];
    if (kk + 16 < K) __builtin_prefetch(wp + 16 * wsK, 0, 1);
    const float* ap = Abase + kk;
    v2f a0, a1, a2, a3;
    a0.x = ap[0];             a0.y = ap[1];
    a1.x = ap[aRowStep];      a1.y = ap[aRowStep + 1];
    a2.x = ap[2 * aRowStep];  a2.y = ap[2 * aRowStep + 1];
    a3.x = ap[3 * aRowStep];  a3.y = ap[3 * aRowStep + 1];
    acc0 = __builtin_amdgcn_wmma_f32_16x16x4_f32(false, a0, false, b, (short)0, acc0, false, false);
    acc1 = __builtin_amdgcn_wmma_f32_16x16x4_f32(false, a1, false, b, (short)0, acc1, false, false);
    acc2 = __builtin_amdgcn_wmma_f32_16x16x4_f32(false, a2, false, b, (short)0, acc2, false, false);
    acc3 = __builtin_amdgcn_wmma_f32_16x16x4_f32(false, a3, false, b, (short)0, acc3, false, false);
  }

  const long cN = n0 + lm;
  float bv = bias ? bias[cN] : 0.0f;
#pragma unroll
  for (int i = 0; i < 8; ++i) {
    const long r = m0 + i + 8 * half;
    float v0 = acc0[i] + bv;
    float v1 = acc1[i] + bv;
    float v2 = acc2[i] + bv;
    float v3 = acc3[i] + bv;
    if (relu) {
      v0 = fmaxf(v0, 0.0f); v1 = fmaxf(v1, 0.0f);
      v2 = fmaxf(v2, 0.0f); v3 = fmaxf(v3, 0.0f);
    }
    C[(r)      * ldc + cN] = v0;
    C[(r + 16) * ldc + cN] = v1;
    C[(r + 32) * ldc + cN] = v2;
    C[(r + 48) * ldc + cN] = v3;
  }
}

// ---------------------------------------------------------------------------
// Radix select (4 x 8-bit MSB-first passes) -> exact key of the 128th-largest
// delta per row + how many elements equal to it are still needed.
// One block (256 thr) per batch row.
// ---------------------------------------------------------------------------
__global__ void topk_select(const float* __restrict__ deltas,
                            unsigned* __restrict__ selmeta, int N, int Kt)
{
  const int b = blockIdx.x;
  const float* row = deltas + (long)b * N;
  __shared__ unsigned hist[256];
  __shared__ unsigned s_prefix, s_remaining;
  if (threadIdx.x == 0) { s_prefix = 0u; s_remaining = (unsigned)Kt; }
  __syncthreads();
  for (int pass = 0; pass < 4; ++pass) {
    const int shift = 24 - pass * 8;
    hist[threadIdx.x] = 0u;
    __syncthreads();
    const unsigned pmask = pass ? (0xFFFFFFFFu << (shift + 8)) : 0u;
    const unsigned pref  = s_prefix;
    const unsigned rem   = s_remaining;
    for (int i = threadIdx.x; i < N; i += 256) {
      const unsigned key = f2key(row[i]);
      if ((key & pmask) == pref) atomicAdd(&hist[(key >> shift) & 0xFF], 1u);
    }
    __syncthreads();
    if (threadIdx.x == 0) {
      unsigned cum = 0;
      for (int bi = 255; bi >= 0; --bi) {
        const unsigned c = hist[bi];
        if (cum + c >= rem) {
          s_remaining = rem - cum;
          s_prefix = pref | ((unsigned)bi << shift);
          break;
        }
        cum += c;
      }
    }
    __syncthreads();
  }
  if (threadIdx.x == 0) {
    selmeta[b * 4 + 0] = s_prefix;     // threshold key (== 128th largest)
    selmeta[b * 4 + 1] = s_remaining;  // # elements equal to key to take
  }
}

__global__ void zero_u32(unsigned* p, int n) {
  const int i = blockIdx.x * blockDim.x + threadIdx.x;
  if (i < n) p[i] = 0u;
}

__global__ void topk_gather(const float* __restrict__ deltas,
                            const unsigned* __restrict__ selmeta,
                            unsigned* __restrict__ counters,   // [B,2]
                            int* __restrict__ idxOut, float* __restrict__ valOut,
                            int N, int Kt)
{
  const int b = blockIdx.y;
  const unsigned thr    = selmeta[b * 4 + 0];
  const unsigned needEq = selmeta[b * 4 + 1];
  const int i = blockIdx.x * blockDim.x + threadIdx.x;
  if (i >= N) return;
  const float f = deltas[(long)b * N + i];
  const unsigned key = f2key(f);
  bool take = false;
  if (key > thr) take = true;
  else if (key == thr) take = (atomicAdd(&counters[b * 2 + 1], 1u) < needEq);
  if (take) {
    const unsigned s = atomicAdd(&counters[b * 2 + 0], 1u);
    if (s < (unsigned)Kt) {
      idxOut[(long)b * Kt + s] = i;
      valOut[(long)b * Kt + s] = f;
    }
  }
}

// meta[b,t,:] = [val, mem_q[idx], mem_t[idx], action[b,0..252]]
__global__ void meta_fill(const int* __restrict__ idx, const float* __restrict__ vals,
                          const float* __restrict__ mem_q, const float* __restrict__ mem_t,
                          const float* __restrict__ action, float* __restrict__ meta)
{
  const long t = (long)blockIdx.x * blockDim.x + threadIdx.x;  // B*KTOP*VSZ
  const int  c = (int)(t & (VSZ - 1));
  const long row = t >> 8;                       // b*KTOP + token
  const long b   = row >> 7;
  float v;
  if      (c == 0) v = vals[row];
  else if (c == 1) v = mem_q[idx[row]];
  else if (c == 2) v = mem_t[idx[row]];
  else             v = action[b * 253 + (c - 3)];
  meta[t] = v;
}

// ---------------------------------------------------------------------------
// LayerNorm over last dim C (optional residual). One block (256 thr) per row.
// In-place safe (each index owned by one thread).
// ---------------------------------------------------------------------------
__global__ void ln_kernel(const float* __restrict__ x, const float* __restrict__ res,
                          const float* __restrict__ g, const float* __restrict__ bta,
                          float* __restrict__ y, int C)
{
  const long row = blockIdx.x;
  const float* xr = x + row * C;
  const float* rr = res ? res + row * C : nullptr;
  __shared__ float red[256];
  float s = 0.0f;
  for (int c = threadIdx.x; c < C; c += 256)
    s += xr[c] + (rr ? rr[c] : 0.0f);
  red[threadIdx.x] = s; __syncthreads();
  for (int off = 128; off > 0; off >>= 1) {
    if ((int)threadIdx.x < off) red[threadIdx.x] += red[threadIdx.x + off];
    __syncthreads();
  }
  const float mean = red[0] / C;
  __syncthreads();
  float s2 = 0.0f;
  for (int c = threadIdx.x; c < C; c += 256) {
    const float d = xr[c] + (rr ? rr[c] : 0.0f) - mean;
    s2 += d * d;
  }
  red[threadIdx.x] = s2; __syncthreads();
  for (int off = 128; off > 0; off >>= 1) {
    if ((int)threadIdx.x < off) red[threadIdx.x] += red[threadIdx.x + off];
    __syncthreads();
  }
  const float inv = rsqrtf(red[0] / C + LN_EPS);
  __syncthreads();
  for (int c = threadIdx.x; c < C; c += 256) {
    const float v = xr[c] + (rr ? rr[c] : 0.0f);
    y[row * C + c] = (v - mean) * inv * g[c] + bta[c];
  }
}

// ---------------------------------------------------------------------------
// Fused attention per batch: S = softmax((q*scale) @ k^T), att = S @ v.
// qkv row layout: [0:32)=q, [32:64)=k, [64:320)=v. 8 waves, WMMA tiles,
// scores staged in 64KB LDS.
// ---------------------------------------------------------------------------
__global__ void attention_kernel(const float* __restrict__ qkv,
                                 float* __restrict__ att)
{
  __shared__ float S[KTOP * KTOP];                 // 64 KB
  const int b    = blockIdx.x;
  const long base = (long)b * KTOP;
  const int wave = threadIdx.x >> 5;
  const int lane = threadIdx.x & 31;
  const int half = lane >> 4;
  const int lm   = lane & 15;
  const int kb   = half * 2;
  const float scale = 0.17677669529663687f;        // 32^-0.5

  // ---- scores: [128,128] = q @ k^T, K-dim = 32 ----
  {
    const int m0 = wave * 16;
    for (int nSub = 0; nSub < 8; ++nSub) {
      const int n0 = nSub * 16;
      v8f acc = {};
      for (int kk = 0; kk < KEYSZ; kk += 4) {
        v2f a, bvec;
        const float* ap = qkv + (base + m0 + lm) * QKVSZ + (kk + kb);
        a.x = ap[0] * scale; a.y = ap[1] * scale;
        const float* bp = qkv + (base + n0 + lm) * QKVSZ + KEYSZ + (kk + kb);
        bvec.x = bp[0]; bvec.y = bp[1];
        acc = __builtin_amdgcn_wmma_f32_16x16x4_f32(false, a, false, bvec,
                                                    (short)0, acc, false, false);
      }
#pragma unroll
      for (int i = 0; i < 8; ++i)
        S[(m0 + i + 8 * half) * KTOP + n0 + lm] = acc[i];
    }
  }
  __syncthreads();

  // ---- softmax rows ----
  if (threadIdx.x < KTOP) {
    float* r = &S[threadIdx.x * KTOP];
    float mx = -3.402823466e38f;
    for (int j = 0; j < KTOP; ++j) mx = fmaxf(mx, r[j]);
    float sum = 0.0f;
    for (int j = 0; j < KTOP; ++j) { const float e = __expf(r[j] - mx); r[j] = e; sum += e; }
    const float inv = 1.0f / sum;
    for (int j = 0; j < KTOP; ++j) r[j] *= inv;
  }
  __syncthreads();

  // ---- att: [128,256] = S @ v, K-dim = 128 ----
  {
    const int m0 = wave * 16;
    for (int nSub = 0; nSub < 16; ++nSub) {
      const int n0 = nSub * 16;
      v8f acc = {};
      for (int kk = 0; kk < KTOP; kk += 4) {
        v2f a, bvec;
        a.x = S[(m0 + lm) * KTOP + kk + kb];
        a.y = S[(m0 + lm) * KTOP + kk + kb + 1];
        bvec.x = qkv[(base + kk + kb)     * QKVSZ + 2 * KEYSZ + n0 + lm];
        bvec.y = qkv[(base + kk + kb + 1) * QKVSZ + 2 * KEYSZ + n0 + lm];
        acc = __builtin_amdgcn_wmma_f32_16x16x4_f32(false, a, false, bvec,
                                                    (short)0, acc, false, false);
      }
#pragma unroll
      for (int i = 0; i < 8; ++i)
        att[(base + m0 + i + 8 * half) * VSZ + n0 + lm] = acc[i];
    }
  }
}

// mean over 128 tokens -> [B, 256]
__global__ void mean_kernel(const float* __restrict__ m2, float* __restrict__ mbar)
{
  const int b = blockIdx.x, c = threadIdx.x;
  float s = 0.0f;
  for (int r = 0; r < KTOP; ++r) s += m2[((long)b * KTOP + r) * VSZ + c];
  mbar[(long)b * VSZ + c] = s * (1.0f / KTOP);
}

// ---------------------------------------------------------------------------
extern "C" void kernel_launch(void* const* d_in, const int* in_sizes, int n_in,
                              void* d_out, int out_size, void* d_ws, size_t ws_size,
                              hipStream_t stream) {
  (void)in_sizes; (void)n_in; (void)out_size; (void)ws_size;
  const float* c      = (const float*)d_in[0];
  const float* action = (const float*)d_in[1];
  const float* mem_c  = (const float*)d_in[2];
  const float* mem_q  = (const float*)d_in[3];
  const float* mem_t  = (const float*)d_in[4];
  const float* W_qkv  = (const float*)d_in[5];
  const float* b_qkv  = (const float*)d_in[6];
  const float* g_ln   = (const float*)d_in[7];
  const float* b_ln   = (const float*)d_in[8];
  const float* g_lnm  = (const float*)d_in[9];
  const float* b_lnm  = (const float*)d_in[10];
  const float* W_mlp1 = (const float*)d_in[11];
  const float* b_mlp1 = (const float*)d_in[12];
  const float* W_mlp2 = (const float*)d_in[13];
  const float* b_mlp2 = (const float*)d_in[14];
  const float* W_p1   = (const float*)d_in[15];
  const float* b_p1   = (const float*)d_in[16];
  const float* W_p2   = (const float*)d_in[17];
  const float* b_p2   = (const float*)d_in[18];
  float* out = (float*)d_out;
  float* ws  = (float*)d_ws;

  // workspace layout (floats); [0,6.4M) first holds deltas, later qkv+att.
  const size_t OFF_DELTAS = 0;            // 6,400,000
  const size_t OFF_QKV    = 0;            // 2,621,440 (deltas dead by then)
  const size_t OFF_ATT    = 2621440;      // 2,097,152 (m1 in-place here)
  const size_t OFF_SEL    = 6400000;      // 1024 u32 (selmeta[64*4] + counters[64*2])
  const size_t OFF_IDX    = 6401024;      // 8192 int
  const size_t OFF_VALS   = 6409216;      // 8192
  const size_t OFF_META   = 6417408;      // 2,097,152
  const size_t OFF_H      = 8514560;      // 2,097,152
  const size_t OFF_MLP    = 10611712;     // 2,097,152 (m2 in-place here)
  const size_t OFF_MBAR   = 12708864;     // 16,384
  const size_t OFF_H1     = 12725248;     // 16,384

  float*    deltas   = ws + OFF_DELTAS;
  unsigned* selmeta  = (unsigned*)(ws + OFF_SEL);
  unsigned* counters = selmeta + 256;
  int*      idx      = (int*)(ws + OFF_IDX);
  float*    vals     = ws + OFF_VALS;
  float*    meta     = ws + OFF_META;
  float*    qkv      = ws + OFF_QKV;
  float*    att      = ws + OFF_ATT;     // becomes m1 in-place
  float*    h        = ws + OFF_H;
  float*    mlp      = ws + OFF_MLP;     // becomes m2 in-place
  float*    mbar     = ws + OFF_MBAR;
  float*    h1       = ws + OFF_H1;

  // 0) zero selection header + counters
  zero_u32<<<4, 256, 0, stream>>>(selmeta, 1024);

  // 1) deltas = c @ mem_c^T   (W strides: wsK=1, wsN=DIM)
  gemm_wmma_f32<<<dim3((NMEM + 127) / 128, 1), 256, 0, stream>>>(
      c, DIM, mem_c, 1, DIM, nullptr, deltas, NMEM, BATCH, NMEM, DIM, 0);

  // 2) exact top-128 threshold per row, then gather set (order-irrelevant:
  //    all downstream token ops are permutation-equivariant + final mean)
  topk_select<<<BATCH, 256, 0, stream>>>(deltas, selmeta, NMEM, KTOP);
  topk_gather<<<dim3((NMEM + 255) / 256, BATCH), 256, 0, stream>>>(
      deltas, selmeta, counters, idx, vals, NMEM, KTOP);

  // 3) meta = [val, mem_q, mem_t, action]
  meta_fill<<<(BATCH * KTOP * VSZ) / 256, 256, 0, stream>>>(
      idx, vals, mem_q, mem_t, action, meta);

  // 4) qkv = LN(meta @ W_qkv + b_qkv)
  gemm_wmma_f32<<<dim3((QKVSZ + 127) / 128, (BATCH * KTOP) / 64), 256, 0, stream>>>(
      meta, VSZ, W_qkv, QKVSZ, 1, b_qkv, qkv, QKVSZ, BATCH * KTOP, QKVSZ, VSZ, 0);
  ln_kernel<<<BATCH * KTOP, 256, 0, stream>>>(qkv, nullptr, g_ln, b_ln, qkv, QKVSZ);

  // 5) attention, then m1 = LN(meta + att) in-place over att
  attention_kernel<<<BATCH, 256, 0, stream>>>(qkv, att);
  ln_kernel<<<BATCH * KTOP, 256, 0, stream>>>(att, meta, g_lnm, b_lnm, att, VSZ);

  // 6) MLP + residual + LN (in-place into mlp)
  gemm_wmma_f32<<<dim3((VSZ + 127) / 128, (BATCH * KTOP) / 64), 256, 0, stream>>>(
      att, VSZ, W_mlp1, VSZ, 1, b_mlp1, h, VSZ, BATCH * KTOP, VSZ, VSZ, 1);
  gemm_wmma_f32<<<dim3((VSZ + 127) / 128, (BATCH * KTOP) / 64), 256, 0, stream>>>(
      h, VSZ, W_mlp2, VSZ, 1, b_mlp2, mlp, VSZ, BATCH * KTOP, VSZ, VSZ, 0);
  ln_kernel<<<BATCH * KTOP, 256, 0, stream>>>(mlp, att, g_lnm, b_lnm, mlp, VSZ);

  // 7) mean over tokens, project out
  mean_kernel<<<BATCH, 256, 0, stream>>>(mlp, mbar);
  gemm_wmma_f32<<<dim3((VSZ + 127) / 128, 1), 256, 0, stream>>>(
      mbar, VSZ, W_p1, VSZ, 1, b_p1, h1, VSZ, BATCH, VSZ, VSZ, 1);
  gemm_wmma_f32<<<dim3((VSZ + 127) / 128, 1), 256, 0, stream>>>(
      h1, VSZ, W_p2, VSZ, 1, b_p2, out, VSZ, BATCH, VSZ, VSZ, 0);
}